// CoxPHLoss_39204461478242
// MI455X (gfx1250) — compile-verified
//
#include <hip/hip_runtime.h>
#include <hip/hip_bf16.h>
#include <stdint.h>

#ifndef __has_builtin
#define __has_builtin(x) 0
#endif

typedef unsigned long long u64;
typedef unsigned int u32;

#define RTILE    4096
#define RTHREADS 256
#define RITEMS   16   // RTILE / RTHREADS

// ---------------------------------------------------------------------------
// Tensor Data Mover helper: 1-D contiguous tile, global -> LDS.
// D# per CDNA5 ISA ch.8: group0 = {count=1 | lds_addr | global_addr | type=2},
// group1 = {data_size | tensor_dim0 | tensor_dim1=1 | tile_dim0 | stride}.
// dsz_code: 0=1B 1=2B 2=4B 3=8B. Issue from one wave; EXEC is ignored by TDM.
// ---------------------------------------------------------------------------
#if defined(__gfx1250__) && __has_builtin(__builtin_amdgcn_tensor_load_to_lds) && \
    __has_builtin(__builtin_amdgcn_s_wait_tensorcnt)
#define HAVE_TDM 1
typedef __attribute__((ext_vector_type(4))) unsigned int v4u_;
typedef __attribute__((ext_vector_type(8))) int v8i_;
typedef __attribute__((ext_vector_type(4))) int v4i_;

__device__ inline void tdm_load_1d(void* lds_dst, const void* gsrc,
                                   unsigned elems, unsigned dsz_code) {
  unsigned long long gaddr = (unsigned long long)(uintptr_t)gsrc;
  unsigned lds_addr = (unsigned)(uintptr_t)lds_dst;  // low 32 bits = LDS offset
  v4u_ g0 = { 1u, lds_addr, (unsigned)(gaddr & 0xFFFFFFFFull),
              (unsigned)(((gaddr >> 32) & 0x1FFFFFFull) | 0x80000000u) };
  v8i_ g1 = { (int)(dsz_code << 16),
              (int)((elems & 0xFFFFu) << 16),               // tensor_dim0 lo
              (int)(((elems >> 16) & 0xFFFFu) | (1u << 16)),// tensor_dim0 hi | tensor_dim1=1
              (int)((elems & 0xFFFFu) << 16),               // tile_dim0
              0,
              (int)elems,                                   // tensor_dim0_stride
              0, 0 };
  v4i_ gz4 = { 0, 0, 0, 0 };
  v8i_ gz8 = { 0, 0, 0, 0, 0, 0, 0, 0 };
  __builtin_amdgcn_tensor_load_to_lds(g0, g1, gz4, gz4, gz8, 0);
}
#endif

// ---------------------------------------------------------------------------
// wave32 / block scan utilities (gfx1250 is wave32: 256 threads = 8 waves)
// ---------------------------------------------------------------------------
template <typename T>
__device__ inline T wave_incl_scan(T v) {
  const int lane = threadIdx.x & 31;
#pragma unroll
  for (int d = 1; d < 32; d <<= 1) {
    T t = __shfl_up(v, d, 32);
    if (lane >= d) v += t;
  }
  return v;
}

// Exclusive block scan over 256 per-thread values. wsum must hold >= 8 T.
// Contains internal barriers; safe to reuse wsum after return.
template <typename T>
__device__ inline T block_excl_scan(T v, T* wsum, T& total) {
  const int lane = threadIdx.x & 31;
  const int wid  = threadIdx.x >> 5;
  T incl = wave_incl_scan(v);
  if (lane == 31) wsum[wid] = incl;
  __syncthreads();
  if (wid == 0) {
    T w = (lane < 8) ? wsum[lane] : T(0);
    T wi = wave_incl_scan(w);
    if (lane < 8) wsum[lane] = wi;
  }
  __syncthreads();
  T base = wid ? wsum[wid - 1] : T(0);
  total = wsum[7];
  __syncthreads();
  return base + incl - v;
}

template <typename T>
__device__ inline T block_reduce_sum(T v, T* wsum) {
#pragma unroll
  for (int d = 16; d; d >>= 1) v += __shfl_down(v, d, 32);
  if ((threadIdx.x & 31) == 0) wsum[threadIdx.x >> 5] = v;
  __syncthreads();
  T r = T(0);
  if (threadIdx.x == 0) {
#pragma unroll
    for (int i = 0; i < 8; ++i) r += wsum[i];
  }
  return r;  // valid on thread 0 only
}

// ---------------------------------------------------------------------------
// 1) Build sortable 64-bit records: (order_key(duration) << 32) | index.
//    Keys are unique -> LSD radix over the top 32 bits reproduces stable argsort.
// ---------------------------------------------------------------------------
__device__ inline u32 float_order_key(float f) {
  u32 k = __float_as_uint(f);
  return (k & 0x80000000u) ? ~k : (k | 0x80000000u);
}

__global__ __launch_bounds__(RTHREADS) void init_keys(const float* __restrict__ dur,
                                                      u64* __restrict__ A, int n) {
  int i = blockIdx.x * RTHREADS + threadIdx.x;
  if (i < n) A[i] = ((u64)float_order_key(dur[i]) << 32) | (u32)i;
}

// ---------------------------------------------------------------------------
// 2a) Per-tile 256-bin histogram, stored bin-major: hist[bin*numTiles + tile]
// ---------------------------------------------------------------------------
__global__ __launch_bounds__(RTHREADS) void radix_hist(const u64* __restrict__ src,
                                                       u32* __restrict__ hist,
                                                       int pass, int n, int numTiles) {
  __shared__ u32 h[256];
  h[threadIdx.x] = 0;
  __syncthreads();
  const int t = blockIdx.x, base = t * RTILE, shift = 32 + 8 * pass;
  for (int k = threadIdx.x; k < RTILE; k += RTHREADS) {
    int g = base + k;
    if (g < n) atomicAdd(&h[(u32)((src[g] >> shift) & 255)], 1u);
  }
  __syncthreads();
  hist[threadIdx.x * numTiles + t] = h[threadIdx.x];
}

// ---------------------------------------------------------------------------
// 2b) In-place exclusive scan of the flat bin-major histogram (single block).
//     Flat order == Σ_{b<bin} total_b + Σ_{t'<t} hist[bin][t'] = scatter base.
// ---------------------------------------------------------------------------
__global__ __launch_bounds__(RTHREADS) void hist_scan(u32* __restrict__ hist, int M) {
  __shared__ u32 wsum[8];
  u32 carry = 0;
  for (int base = 0; base < M; base += RTILE) {
    u32 v[RITEMS], s = 0;
    const int lo = base + threadIdx.x * RITEMS;
#pragma unroll
    for (int k = 0; k < RITEMS; ++k) {
      u32 x = (lo + k < M) ? hist[lo + k] : 0u;
      v[k] = x; s += x;
    }
    u32 total, ex = block_excl_scan(s, wsum, total);
    u32 run = carry + ex;
#pragma unroll
    for (int k = 0; k < RITEMS; ++k) {
      if (lo + k < M) hist[lo + k] = run;
      run += v[k];
    }
    carry += total;
    __syncthreads();
  }
}

// ---------------------------------------------------------------------------
// 2c) Stable scatter. Tile staged into LDS by the Tensor Data Mover (32 KB,
//     one wave-level DMA, no VGPR staging); local stable sort by the 8-bit
//     digit via 8 one-bit splits; coalesced global scatter via scanned bases.
// ---------------------------------------------------------------------------
__global__ __launch_bounds__(RTHREADS) void radix_scatter(const u64* __restrict__ src,
                                                          u64* __restrict__ dst,
                                                          const u32* __restrict__ hist,
                                                          int pass, int n, int numTiles) {
  __shared__ u64 tile[RTILE];            // 32 KB of the 320 KB/WGP LDS
  __shared__ u32 wsum[8];
  __shared__ u32 binStart[256];
  const int t = blockIdx.x, base = t * RTILE, shift = 32 + 8 * pass;
  const int cnt = min(RTILE, n - base);

#ifdef HAVE_TDM
  if (threadIdx.x < 32) tdm_load_1d(&tile[0], src + base, (unsigned)cnt, 3u /*8B*/);
  // pad the (disjoint) LDS tail while the DMA is in flight
  for (int k = cnt + threadIdx.x; k < RTILE; k += RTHREADS) tile[k] = ~0ull;
  if (threadIdx.x < 32) __builtin_amdgcn_s_wait_tensorcnt(0);
  __syncthreads();
#else
  for (int k = threadIdx.x; k < RTILE; k += RTHREADS) {
    int g = base + k;
    tile[k] = (g < n) ? src[g] : ~0ull;  // pad keys sort to the end
  }
  __syncthreads();
#endif

  for (int b = 0; b < 8; ++b) {
    u64 v[RITEMS];
    u32 zeros = 0;
    const int lo = threadIdx.x * RITEMS;
#pragma unroll
    for (int k = 0; k < RITEMS; ++k) {
      v[k] = tile[lo + k];
      zeros += 1u - (u32)((v[k] >> (shift + b)) & 1ull);
    }
    // block_excl_scan's first barrier guarantees all reads precede any write
    u32 total, zbase = block_excl_scan(zeros, wsum, total);
    u32 zrun = zbase;
#pragma unroll
    for (int k = 0; k < RITEMS; ++k) {
      u32 p   = (u32)(lo + k);
      u32 isz = 1u - (u32)((v[k] >> (shift + b)) & 1ull);
      u32 d   = isz ? zrun : (total + p - zrun);  // stable 1-bit partition
      tile[d] = v[k];
      zrun += isz;
    }
    __syncthreads();
  }

  for (int k = threadIdx.x; k < RTILE; k += RTHREADS) {
    u32 bin  = (u32)((tile[k] >> shift) & 255);
    u32 prev = (k == 0) ? 256u : (u32)((tile[k - 1] >> shift) & 255);
    if (bin != prev) binStart[bin] = (u32)k;
  }
  __syncthreads();

  for (int k = threadIdx.x; k < RTILE; k += RTHREADS) {
    u64 v = tile[k];
    if (v == ~0ull) continue;  // padding
    u32 bin = (u32)((v >> shift) & 255);
    dst[hist[bin * numTiles + t] + ((u32)k - binStart[bin])] = v;
  }
}

// ---------------------------------------------------------------------------
// 3) Global max of log_h (gamma), deterministic two-level tree.
// ---------------------------------------------------------------------------
__global__ __launch_bounds__(RTHREADS) void max_part(const float* __restrict__ x,
                                                     float* __restrict__ part, int n) {
  __shared__ float w[8];
  float m = -__builtin_inff();
  for (int i = blockIdx.x * RTHREADS + threadIdx.x; i < n; i += gridDim.x * RTHREADS)
    m = fmaxf(m, x[i]);
#pragma unroll
  for (int d = 16; d; d >>= 1) m = fmaxf(m, __shfl_down(m, d, 32));
  if ((threadIdx.x & 31) == 0) w[threadIdx.x >> 5] = m;
  __syncthreads();
  if (threadIdx.x == 0) {
    float r = w[0];
#pragma unroll
    for (int i = 1; i < 8; ++i) r = fmaxf(r, w[i]);
    part[blockIdx.x] = r;
  }
}

__global__ __launch_bounds__(RTHREADS) void max_final(const float* __restrict__ part,
                                                      int m, float* __restrict__ gamma) {
  __shared__ float w[8];
  float v = -__builtin_inff();
  for (int i = threadIdx.x; i < m; i += RTHREADS) v = fmaxf(v, part[i]);
#pragma unroll
  for (int d = 16; d; d >>= 1) v = fmaxf(v, __shfl_down(v, d, 32));
  if ((threadIdx.x & 31) == 0) w[threadIdx.x >> 5] = v;
  __syncthreads();
  if (threadIdx.x == 0) {
    float r = w[0];
#pragma unroll
    for (int i = 1; i < 8; ++i) r = fmaxf(r, w[i]);
    gamma[0] = r;
  }
}

// ---------------------------------------------------------------------------
// 4) Fused: e[i] = exp(log_h[order[i]] - gamma) (stored), tile sum reduced
//    in the same pass (saves a full re-read of e).
// ---------------------------------------------------------------------------
__global__ __launch_bounds__(RTHREADS) void exp_gather_tilesum(
    const u64* __restrict__ sorted, const float* __restrict__ log_h,
    const float* __restrict__ gamma, float* __restrict__ e,
    float* __restrict__ tsum, int n) {
  __shared__ float w[8];
  const int t = blockIdx.x, base = t * RTILE;
  const float g = gamma[0];
  float s = 0.f;
  for (int k = threadIdx.x; k < RTILE; k += RTHREADS) {
    int i = base + k;
    if (i < n) {
      u32 idx = (u32)(sorted[i] & 0xFFFFFFFFull);
      float ev = expf(log_h[idx] - g);
      e[i] = ev;
      s += ev;
    }
  }
  float r = block_reduce_sum(s, w);
  if (threadIdx.x == 0) tsum[t] = r;
}

// tileSuffix[t] = sum_{t' > t} tsum[t']  (exclusive suffix), via reversed fwd scan
__global__ __launch_bounds__(RTHREADS) void suffix_scan(const float* __restrict__ tsum,
                                                        float* __restrict__ tsuf, int m) {
  __shared__ float wsum[8];
  float carry = 0.f;
  for (int base = 0; base < m; base += RTILE) {
    float v[RITEMS], s = 0.f;
    const int lo = base + threadIdx.x * RITEMS;
#pragma unroll
    for (int k = 0; k < RITEMS; ++k) {
      int j = lo + k;
      float x = (j < m) ? tsum[m - 1 - j] : 0.f;
      v[k] = x; s += x;
    }
    float total, ex = block_excl_scan(s, wsum, total);
    float run = carry + ex;
#pragma unroll
    for (int k = 0; k < RITEMS; ++k) {
      int j = lo + k;
      if (j < m) tsuf[m - 1 - j] = run;
      run += v[k];
    }
    carry += total;
    __syncthreads();
  }
}

// ---------------------------------------------------------------------------
// 5) Fused: stage e-tile into LDS with the Tensor Data Mover, in-tile
//    inclusive suffix, lcse = log(s)+gamma, event-masked term reduction.
// ---------------------------------------------------------------------------
__global__ __launch_bounds__(RTHREADS) void loss_tile(const u64* __restrict__ sorted,
                                                      const float* __restrict__ log_h,
                                                      const int* __restrict__ events,
                                                      const float* __restrict__ e,
                                                      const float* __restrict__ tsuf,
                                                      const float* __restrict__ gamma,
                                                      float* __restrict__ ploss, int n) {
  __shared__ float ef[RTILE];  // 16 KB tile staged from global
  __shared__ float wsum[8];
  const int t = blockIdx.x, base = t * RTILE;
  const int cnt = min(RTILE, n - base);

#ifdef HAVE_TDM
  if (threadIdx.x < 32) {
    tdm_load_1d(&ef[0], e + base, (unsigned)cnt, 2u /*4B*/);
    __builtin_amdgcn_s_wait_tensorcnt(0);
  }
  __syncthreads();
#else
  for (int k = threadIdx.x; k < cnt; k += RTHREADS) ef[k] = e[base + k];
  __syncthreads();
#endif

  const float g = gamma[0];
  const float tail = tsuf[t];
  const int lo = threadIdx.x * RITEMS;

  float v[RITEMS], s = 0.f;
#pragma unroll
  for (int k = 0; k < RITEMS; ++k) {
    int j = lo + k;
    float x = (j < cnt) ? ef[j] : 0.f;
    v[k] = x; s += x;
  }
  float total, ex = block_excl_scan(s, wsum, total);

  float loss = 0.f;
  float run = ex;  // exclusive prefix of e within tile at position lo+k
#pragma unroll
  for (int k = 0; k < RITEMS; ++k) {
    int j = lo + k;
    if (j < cnt) {
      u32 idx = (u32)(sorted[base + j] & 0xFFFFFFFFull);
      float sfx = tail + (total - run);       // inclusive suffix sum at j
      if (events[idx]) loss += log_h[idx] - (logf(sfx) + g);
    }
    run += v[k];
  }
  __syncthreads();
  float r = block_reduce_sum(loss, wsum);
  if (threadIdx.x == 0) ploss[t] = r;
}

__global__ __launch_bounds__(RTHREADS) void final_loss(const float* __restrict__ ploss,
                                                       int m, float* __restrict__ out) {
  __shared__ float w[8];
  float s = 0.f;
  for (int i = threadIdx.x; i < m; i += RTHREADS) s += ploss[i];
  float r = block_reduce_sum(s, w);
  if (threadIdx.x == 0) out[0] = -r;
}

// ---------------------------------------------------------------------------
// Launcher
// ---------------------------------------------------------------------------
extern "C" void kernel_launch(void* const* d_in, const int* in_sizes, int n_in,
                              void* d_out, int out_size, void* d_ws, size_t ws_size,
                              hipStream_t stream) {
  const float* log_h = (const float*)d_in[0];
  const float* dur   = (const float*)d_in[1];
  const int*   ev    = (const int*)d_in[2];
  const int n = in_sizes[0];
  const int numTiles = (n + RTILE - 1) / RTILE;

  // Workspace layout (n = 2^23 -> ~130 MB):
  //   A: n u64 | B: n u64 (reused as e after sort) | hist: 256*numTiles u32 | small
  char* ws = (char*)d_ws;
  u64* A = (u64*)ws;
  u64* B = (u64*)(ws + (size_t)n * 8);
  u32* hist = (u32*)(ws + (size_t)n * 16);
  char* small = ws + (size_t)n * 16 + (size_t)256 * numTiles * 4;
  float* tsum  = (float*)small;
  float* tsufA = tsum + numTiles;
  float* ploss = tsufA + numTiles;
  float* pmax  = ploss + numTiles;   // 512
  float* gamma = pmax + 512;
  float* e = (float*)B;              // reuse B after the final radix pass lands in A

  const int lin = (n + RTHREADS - 1) / RTHREADS;

  init_keys<<<lin, RTHREADS, 0, stream>>>(dur, A, n);

  for (int pass = 0; pass < 4; ++pass) {   // A->B->A->B->A : result in A
    const u64* src = (pass & 1) ? B : A;
    u64*       dst = (pass & 1) ? A : B;
    radix_hist<<<numTiles, RTHREADS, 0, stream>>>(src, hist, pass, n, numTiles);
    hist_scan<<<1, RTHREADS, 0, stream>>>(hist, 256 * numTiles);
    radix_scatter<<<numTiles, RTHREADS, 0, stream>>>(src, dst, hist, pass, n, numTiles);
  }

  max_part<<<512, RTHREADS, 0, stream>>>(log_h, pmax, n);
  max_final<<<1, RTHREADS, 0, stream>>>(pmax, 512, gamma);
  exp_gather_tilesum<<<numTiles, RTHREADS, 0, stream>>>(A, log_h, gamma, e, tsum, n);
  suffix_scan<<<1, RTHREADS, 0, stream>>>(tsum, tsufA, numTiles);
  loss_tile<<<numTiles, RTHREADS, 0, stream>>>(A, log_h, ev, e, tsufA, gamma, ploss, n);
  final_loss<<<1, RTHREADS, 0, stream>>>(ploss, numTiles, (float*)d_out);
}